// PositionalEncoding_30331059044545
// MI455X (gfx1250) — compile-verified
//
#include <hip/hip_runtime.h>

// PositionalEncoding: out[i,:] = x[i,:] + pos_embedding[i - seg_start(batch[i]), :]
//   N = 200000 rows, H = 512, table = 1000 x 512 f32 (2 MB -> L2-resident on MI455X).
// Pure streaming: ~0.82 GB HBM traffic -> ~35 us floor at 23.3 TB/s. No matmul,
// so WMMA is inapplicable; the optimization targets are:
//   - b128 vector loads/stores (one float4 per lane, 128 lanes per row)
//   - 4 rows per block: 8 independent global_load_b128 in flight per lane
//     (memory-level parallelism), metadata amortized, 50k workgroups not 200k
//   - block-uniform row indices -> batch[row]/seg_start[g] become s_load_b32
//   - NT temporal hints on streamed x/out so the embedding table owns L2
// (Hot kernel defined FIRST so the disasm snippet shows its codegen.)

typedef __attribute__((ext_vector_type(4))) float v4f;

#define HDIM 512
#define VEC 4
#define ROW_THREADS (HDIM / VEC)   // 128 threads = 4 wave32, one float4/lane/row
#define ROWS_PER_BLOCK 4

// Hot kernel: 4 consecutive rows per block. All row indices are block-uniform so
// the per-row metadata lowers to scalar loads; each lane then issues 8
// independent b128 loads (4 NT x + 4 pe) before the adds, maximizing MLP.
__global__ void __launch_bounds__(ROW_THREADS)
posenc_add_kernel(const float* __restrict__ x,
                  const int* __restrict__ batch,
                  const float* __restrict__ pe,
                  const int* __restrict__ seg_start,
                  float* __restrict__ out,
                  int n) {
    const int row0 = blockIdx.x * ROWS_PER_BLOCK;   // uniform
    const int col  = threadIdx.x * VEC;

    v4f xv[ROWS_PER_BLOCK];
    v4f pv[ROWS_PER_BLOCK];

#pragma unroll
    for (int r = 0; r < ROWS_PER_BLOCK; ++r) {
        const int row = row0 + r;
        if (row < n) {                              // uniform guard
            const int g   = batch[row];             // -> s_load_b32
            const int pos = row - seg_start[g];     // -> s_load_b32
            const v4f* xp = (const v4f*)(x  + (size_t)row * HDIM + col);
            const v4f* pp = (const v4f*)(pe + (size_t)pos * HDIM + col);
            xv[r] = __builtin_nontemporal_load(xp); // streamed once: NT
            pv[r] = *pp;                            // 2 MB table: temporal, L2 hit
        }
    }

#pragma unroll
    for (int r = 0; r < ROWS_PER_BLOCK; ++r) {
        const int row = row0 + r;
        if (row < n) {
            v4f* op = (v4f*)(out + (size_t)row * HDIM + col);
            __builtin_nontemporal_store(xv[r] + pv[r], op); // streamed once: NT
        }
    }
}

// Boundary kernel: batch is sorted; i is a segment start iff i==0 or
// batch[i]!=batch[i-1]. Scatter start index into seg_start[graph_id]. Each
// present id written exactly once per launch (no race); only present ids are
// ever read back.
__global__ void __launch_bounds__(256)
seg_start_kernel(const int* __restrict__ batch, int* __restrict__ seg_start, int n) {
    int i = blockIdx.x * blockDim.x + threadIdx.x;
    if (i >= n) return;
    int b = batch[i];
    if (i == 0 || batch[i - 1] != b) {
        seg_start[b] = i;
    }
}

extern "C" void kernel_launch(void* const* d_in, const int* in_sizes, int n_in,
                              void* d_out, int out_size, void* d_ws, size_t ws_size,
                              hipStream_t stream) {
    const float* x     = (const float*)d_in[0];   // [N, 512] f32
    const int*   batch = (const int*)  d_in[1];   // [N] sorted segment ids
    const float* pe    = (const float*)d_in[2];   // [1000, 512] f32
    float*       out   = (float*)d_out;

    const int n = in_sizes[1];                    // N = number of rows
    int* seg_start = (int*)d_ws;                  // <= 1000 ints of scratch

    const int t1 = 256;
    seg_start_kernel<<<(n + t1 - 1) / t1, t1, 0, stream>>>(batch, seg_start, n);

    const int nblk = (n + ROWS_PER_BLOCK - 1) / ROWS_PER_BLOCK;
    posenc_add_kernel<<<nblk, ROW_THREADS, 0, stream>>>(x, batch, pe, seg_start, out, n);
}